// Decoder_60129542493
// MI455X (gfx1250) — compile-verified
//
#include <hip/hip_runtime.h>
#include <hip/hip_bf16.h>
#include <stdint.h>

#define B_DIM 256
#define O_DIM 128
#define D_DIM 128
#define IN_DIM (O_DIM * D_DIM)   // 16384
#define KSPLIT 4
#define KCH (IN_DIM / KSPLIT)    // 4096
#define NEG_MASK -9.0e15f
#define WT_STRIDE 132            // 128 + TDM pad (4 dwords) -> conflict-light LDS rows

typedef float v2f __attribute__((ext_vector_type(2)));
typedef float v8f __attribute__((ext_vector_type(8)));
typedef unsigned int v4u __attribute__((ext_vector_type(4)));
typedef int v8i __attribute__((ext_vector_type(8)));
typedef int v4i __attribute__((ext_vector_type(4)));

#if __has_builtin(__builtin_amdgcn_tensor_load_to_lds)
#define USE_TDM 1
#else
#define USE_TDM 0
#endif

// ---------------------------------------------------------------------------
// Kernel 1: K-split partial GEMM. p0part[kc][256][128] = enc_kc @ W_kc^T.
// f32 WMMA 16x16x4; one 16x16 tile per wave; 8 waves/block = 128 cols;
// blockIdx = mtile (16) x kchunk (4). Deterministic: no atomics, partials
// reduced in fixed order by the decode kernel.
// ---------------------------------------------------------------------------
__global__ __launch_bounds__(256) void p0_gemm_wmma(
    const float* __restrict__ enc, const float* __restrict__ W,
    float* __restrict__ p0part) {
  const int wave  = threadIdx.x >> 5;
  const int lane  = threadIdx.x & 31;
  const int mtile = blockIdx.x & 15;
  const int kc    = blockIdx.x >> 4;        // 0..3
  const int m0    = mtile << 4;
  const int n0    = wave << 4;
  const int half  = lane >> 4;              // 0 or 1
  const int l16   = lane & 15;
  const int kbase = kc * KCH;
  // A lane map (16x4 f32): row M = l16, VGPR{0,1} = K{2*half, 2*half+1}
  // B lane map (4x16 f32): col N = l16, same K split
  const float* arow = enc + (size_t)(m0 + l16) * IN_DIM + kbase + (half << 1);
  const float* brow = W   + (size_t)(n0 + l16) * IN_DIM + kbase + (half << 1);
  v8f acc = {0.f, 0.f, 0.f, 0.f, 0.f, 0.f, 0.f, 0.f};
  for (int k = 0; k < KCH; k += 16) {
#pragma unroll
    for (int kk = 0; kk < 16; kk += 4) {
      v2f a = *(const v2f*)(arow + k + kk);
      v2f b = *(const v2f*)(brow + k + kk);
      acc = __builtin_amdgcn_wmma_f32_16x16x4_f32(false, a, false, b,
                                                  (short)0, acc, false, false);
    }
  }
  const int rbase = m0 + (half << 3);
  float* dst = p0part + ((size_t)kc * B_DIM + rbase) * O_DIM + n0 + l16;
#pragma unroll
  for (int i = 0; i < 8; ++i)
    dst[(size_t)i * O_DIM] = acc[i];
}

// ---------------------------------------------------------------------------
// threefry2x32 (JAX default PRNG), 20 rounds
// ---------------------------------------------------------------------------
__device__ __forceinline__ unsigned rotl32(unsigned v, int r) {
  return (v << r) | (v >> (32 - r));
}

__device__ __forceinline__ void threefry2x32(unsigned key0, unsigned key1,
                                             unsigned x0, unsigned x1,
                                             unsigned& out0, unsigned& out1) {
  unsigned ks[3];
  ks[0] = key0; ks[1] = key1; ks[2] = key0 ^ key1 ^ 0x1BD11BDAu;
  x0 += ks[0]; x1 += ks[1];
  const int rotA[4] = {13, 15, 26, 6};
  const int rotB[4] = {17, 29, 16, 24};
#pragma unroll
  for (int i = 0; i < 5; ++i) {
#pragma unroll
    for (int j = 0; j < 4; ++j) {
      int r = (i & 1) ? rotB[j] : rotA[j];
      x0 += x1; x1 = rotl32(x1, r); x1 ^= x0;
    }
    x0 += ks[(i + 1) % 3];
    x1 += ks[(i + 2) % 3] + (unsigned)(i + 1);
  }
  out0 = x0; out1 = x1;
}

__device__ __forceinline__ float gumbel_from_bits(unsigned bits) {
  const float tiny = 1.175494350822288e-38f;  // FLT_MIN
  unsigned fb = (bits >> 9) | 0x3f800000u;
  float f = __uint_as_float(fb) - 1.0f;       // [0,1)
  float u = f * (1.0f - tiny) + tiny;
  u = fmaxf(tiny, u);
  return -logf(-logf(u));
}

// ---------------------------------------------------------------------------
// Kernel 2: per-row 128-step decode loop. grid=256 rows, block=128 threads.
// W block for the incremental update is DMA'd into LDS by the Tensor Data
// Mover each step (with padded rows), falling back to direct global reads.
// ---------------------------------------------------------------------------
__global__ __launch_bounds__(128) void decode_loop(
    const float* __restrict__ enc,     // [256,128,128]
    const float* __restrict__ W,       // [128,16384]
    const float* __restrict__ p0part,  // [4,256,128]
    const float* __restrict__ bias,    // [128]
    float* __restrict__ out_pos,       // [256*128]
    float* __restrict__ out_ls,        // [256]
    float* __restrict__ out_err) {     // [256]
  const int b   = blockIdx.x;
  const int tid = threadIdx.x;

  extern __shared__ float Wt[];        // [128][132] dynamic, 67584 B
  __shared__ float    p_s[O_DIM];
  __shared__ float    mask_s[O_DIM];
  __shared__ float    red_s[O_DIM];
  __shared__ int      ired_s[O_DIM];
  __shared__ float    logp_s[O_DIM];
  __shared__ float    encv_s[D_DIM];
  __shared__ unsigned keybits[2 * O_DIM];
  __shared__ int      top_s;

  // fold K-split partials in fixed order (deterministic) + bias
  {
    const size_t idx = (size_t)b * O_DIM + tid;
    float v = (p0part[idx] + p0part[(size_t)B_DIM * O_DIM + idx]) +
              (p0part[2 * (size_t)B_DIM * O_DIM + idx] +
               p0part[3 * (size_t)B_DIM * O_DIM + idx]);
    p_s[tid] = v + bias[tid];
  }
  mask_s[tid] = 1.0f;
  // keys = jax.random.split(jax.random.key(42), 128):
  // threefry(key=(0,42), counts=iota(256)) paired as (t, t+128).
  {
    unsigned o0, o1;
    threefry2x32(0u, 42u, (unsigned)tid, (unsigned)(tid + 128), o0, o1);
    keybits[tid]       = o0;
    keybits[tid + 128] = o1;
  }
  float lsAcc = 0.0f;
  int   errAcc = 0;
  __syncthreads();

  for (int j = 0; j < O_DIM; ++j) {
    const float myp = p_s[tid];

    // ---- 1) rank-j element of raw p (stable descending argsort semantics)
    int cnt = 0;
#pragma unroll 8
    for (int o = 0; o < O_DIM; ++o) {
      float q = p_s[o];
      cnt += (q > myp) || ((q == myp) && (o < tid));
    }
    if (cnt == j) top_s = tid;   // exactly one thread (total order)
    __syncthreads();

    // ---- 2) masked log-softmax
    const float pm = (mask_s[tid] != 0.0f) ? myp : NEG_MASK;
    red_s[tid] = pm;
    __syncthreads();
#pragma unroll
    for (int s = 64; s > 0; s >>= 1) {
      if (tid < s) red_s[tid] = fmaxf(red_s[tid], red_s[tid + s]);
      __syncthreads();
    }
    const float mx = red_s[0];
    __syncthreads();
    const float sh = pm - mx;
    red_s[tid] = expf(sh);
    __syncthreads();
#pragma unroll
    for (int s = 64; s > 0; s >>= 1) {
      if (tid < s) red_s[tid] += red_s[tid + s];
      __syncthreads();
    }
    const float lse = logf(red_s[0]);
    __syncthreads();
    const float lp = sh - lse;
    logp_s[tid] = lp;

    // ---- 3) gumbel sample: argmax(logp + g), first index on ties
    const unsigned i    = (unsigned)(b * O_DIM + tid);
    const unsigned lo   = i & 16383u;
    const unsigned hsel = i >> 14;
    unsigned g0, g1;
    threefry2x32(keybits[2 * j], keybits[2 * j + 1], lo, lo + 16384u, g0, g1);
    const float g   = gumbel_from_bits(hsel ? g1 : g0);
    const float val = lp + g;
    red_s[tid]  = val;
    ired_s[tid] = tid;
    __syncthreads();
#pragma unroll
    for (int s = 64; s > 0; s >>= 1) {
      if (tid < s) {
        float a = red_s[tid], c = red_s[tid + s];
        int   ia = ired_s[tid], ic = ired_s[tid + s];
        if (c > a || (c == a && ic < ia)) { red_s[tid] = c; ired_s[tid] = ic; }
      }
      __syncthreads();
    }
    const int pos = ired_s[0];

    if (tid == 0) {
      errAcc += (mask_s[top_s] == 0.0f) ? 1 : 0;   // mask BEFORE this step's update
      lsAcc  += logp_s[pos];
      out_pos[b * O_DIM + (O_DIM - 1 - j)] = (float)pos;  // prepend order
    }
    __syncthreads();   // err read of mask must precede mask update

    // ---- 4) update mask and incrementally update p for next step
    if (tid == pos) mask_s[tid] = 0.0f;
    if (j < O_DIM - 1) {
      encv_s[tid] = enc[((size_t)b * O_DIM + pos) * D_DIM + tid];
#if USE_TDM
      // Tensor Data Mover: DMA the 128x128 f32 W block (cols pos*128..+127)
      // into LDS with 4-dword row padding (row stride = 132 floats).
      if (tid < 32) {  // wave 0 issues; EXEC ignored by TDM, branch is uniform
        unsigned long long ga =
            (unsigned long long)(uintptr_t)W + (unsigned)(pos << 9);
        unsigned ldsb = (unsigned)(uintptr_t)(void*)Wt;
        // D# group0: count=1 | lds_addr | global_addr | type=2
        unsigned d0 = 1u;
        unsigned d1 = __builtin_amdgcn_readfirstlane(ldsb);
        unsigned d2 = __builtin_amdgcn_readfirstlane((unsigned)ga);
        unsigned d3 = (__builtin_amdgcn_readfirstlane(
                           (unsigned)((ga >> 32) & 0x1FFFFFFull))) |
                      0x80000000u;  // type=2 ("image")
        v4u g0v = {d0, d1, d2, d3};
        // D# group1:
        // w0: data_size=4B(2<<16) | pad_enable(1<<20) | pad_interval=6
        //     (128 dwords, 6<<22) | pad_amount=3 (4 dwords, 3<<25)
        // w1: tensor_dim0=16384 low16 in [63:48]
        // w2: tensor_dim0 hi=0, tensor_dim1=128 low16 in [31:16]
        // w3: tensor_dim1 hi=0, tile_dim0=128 in [127:112]
        // w4: tile_dim1=128 in [143:128], tile_dim2=0
        // w5: tensor_dim0_stride=16384 (low32), w6/w7: 0
        v8i g1v = {(int)0x07920000, (int)0x40000000, (int)0x00800000,
                   (int)0x00800000, (int)0x00000080, (int)16384, 0, 0};
        v4i gz4 = {0, 0, 0, 0};
        v8i gz8 = {0, 0, 0, 0, 0, 0, 0, 0};
        __builtin_amdgcn_tensor_load_to_lds(g0v, g1v, gz4, gz4, gz8, 0);
        __builtin_amdgcn_s_wait_tensorcnt(0);
      }
      __syncthreads();
      const float* wr = &Wt[tid * WT_STRIDE];
      float acc = 0.0f;
#pragma unroll 8
      for (int d = 0; d < D_DIM; d += 4) {
        float4 wv = *(const float4*)(wr + d);   // 16B aligned: 132*4 % 16 == 0
        acc = fmaf(encv_s[d],     wv.x, acc);
        acc = fmaf(encv_s[d + 1], wv.y, acc);
        acc = fmaf(encv_s[d + 2], wv.z, acc);
        acc = fmaf(encv_s[d + 3], wv.w, acc);
      }
#else
      __syncthreads();
      const float* wrow = W + (size_t)tid * IN_DIM + pos * D_DIM;
      float acc = 0.0f;
#pragma unroll 8
      for (int d = 0; d < D_DIM; d += 4) {
        float4 wv = *(const float4*)(wrow + d);
        acc = fmaf(encv_s[d],     wv.x, acc);
        acc = fmaf(encv_s[d + 1], wv.y, acc);
        acc = fmaf(encv_s[d + 2], wv.z, acc);
        acc = fmaf(encv_s[d + 3], wv.w, acc);
      }
#endif
      p_s[tid] = myp - acc;
    }
    __syncthreads();
  }

  if (tid == 0) {
    out_ls[b]  = lsAcc;
    out_err[b] = (float)errAcc;
  }
}

// ---------------------------------------------------------------------------
extern "C" void kernel_launch(void* const* d_in, const int* in_sizes, int n_in,
                              void* d_out, int out_size, void* d_ws, size_t ws_size,
                              hipStream_t stream) {
  const float* enc  = (const float*)d_in[0];   // [256,128,128]
  const float* W    = (const float*)d_in[1];   // [128,16384]
  const float* bias = (const float*)d_in[2];   // [128]
  float* out    = (float*)d_out;               // positions | log_softmaxs | errors
  float* p0part = (float*)d_ws;                // [4,256,128] scratch (512 KB)

  p0_gemm_wmma<<<16 * KSPLIT, 256, 0, stream>>>(enc, W, p0part);

  const size_t dyn_lds = (size_t)O_DIM * WT_STRIDE * sizeof(float);  // 67584 B
  decode_loop<<<B_DIM, 128, dyn_lds, stream>>>(
      enc, W, p0part, bias,
      out,                                  // 256*128 positions
      out + B_DIM * O_DIM,                  // 256 log-softmax sums
      out + B_DIM * O_DIM + B_DIM);         // 256 error counts
}